// PointTransformerSeg_16750372454758
// MI455X (gfx1250) — compile-verified
//
#include <hip/hip_runtime.h>
#include <math.h>

// Problem constants (from reference)
#define NPTS 65536
#define NS   16
#define C    64
#define CH   8            // MID // S
#define PAIRS (NPTS*NS)   // 1,048,576
#define EPSF 1e-5f

typedef __attribute__((ext_vector_type(2))) float v2f;
typedef __attribute__((ext_vector_type(8))) float v8f;

// ---------------------------------------------------------------------------
// BN params for the 3-channel positional branch
// ---------------------------------------------------------------------------
struct BNp { float m0,m1,m2, s0,s1,s2, b0,b1,b2; };

__device__ __forceinline__ BNp make_bnp(const float* __restrict__ ps,
                                        const float* __restrict__ gp,
                                        const float* __restrict__ bp) {
  const float invM = 1.0f / (float)PAIRS;
  BNp r;
  float m0 = ps[0]*invM, m1 = ps[1]*invM, m2 = ps[2]*invM;
  float v0 = ps[3]*invM - m0*m0;
  float v1 = ps[4]*invM - m1*m1;
  float v2 = ps[5]*invM - m2*m2;
  r.m0=m0; r.m1=m1; r.m2=m2;
  r.s0 = rsqrtf(v0+EPSF)*gp[0];
  r.s1 = rsqrtf(v1+EPSF)*gp[1];
  r.s2 = rsqrtf(v2+EPSF)*gp[2];
  r.b0 = bp[0]; r.b1 = bp[1]; r.b2 = bp[2];
  return r;
}

// relu(BN(rel_p @ Wp1 + bp1)) -> (a0,a1,a2)
__device__ __forceinline__ void pfeat(const float* __restrict__ p, int n, int id,
                                      const float* __restrict__ Wp1,
                                      const float* __restrict__ bp1,
                                      const BNp& B, float& a0, float& a1, float& a2) {
  float r0 = p[id*3+0]-p[n*3+0];
  float r1 = p[id*3+1]-p[n*3+1];
  float r2 = p[id*3+2]-p[n*3+2];
  float t0 = fmaf(r2, Wp1[6], fmaf(r1, Wp1[3], fmaf(r0, Wp1[0], bp1[0])));
  float t1 = fmaf(r2, Wp1[7], fmaf(r1, Wp1[4], fmaf(r0, Wp1[1], bp1[1])));
  float t2 = fmaf(r2, Wp1[8], fmaf(r1, Wp1[5], fmaf(r0, Wp1[2], bp1[2])));
  a0 = fmaxf(fmaf(t0-B.m0, B.s0, B.b0), 0.f);
  a1 = fmaxf(fmaf(t1-B.m1, B.s1, B.b1), 0.f);
  a2 = fmaxf(fmaf(t2-B.m2, B.s2, B.b2), 0.f);
}

// ---------------------------------------------------------------------------
// Kernel 1: xq/xk/xv = x @ W + b via V_WMMA_F32_16X16X4_F32.
// Each wave owns one of the 12 (matrix, N-tile) combos; the B fragment
// (32 floats) + bias are hoisted into registers once, then the wave
// grid-strides over M-tiles: 16 b64 A-loads + 16 WMMAs + 8 stores per tile.
// A (16x4 f32): lanes 0-15 rows M, VGPR0=K0 VGPR1=K1; lanes 16-31 K2/K3.
// B (4x16 f32): lanes hold N, VGPR0=K0(|K2), VGPR1=K1(|K3).
// ---------------------------------------------------------------------------
#define QKV_WAVES 12288   // 12 combos x 1024 waves -> 4 M-tiles per wave

__global__ void __launch_bounds__(256) k_qkv(
    const float* __restrict__ x,
    const float* __restrict__ Wq, const float* __restrict__ bq,
    const float* __restrict__ Wk, const float* __restrict__ bk,
    const float* __restrict__ Wv, const float* __restrict__ bv,
    float* __restrict__ xq, float* __restrict__ xk, float* __restrict__ xv)
{
  const int wave  = (blockIdx.x * blockDim.x + threadIdx.x) >> 5;
  const int lane  = threadIdx.x & 31;
  const int combo = wave % 12;
  const int mat   = combo % 3;
  const int nt    = combo / 3;          // 0..3
  const float* W    = (mat==0) ? Wq : (mat==1) ? Wk : Wv;
  const float* bias = (mat==0) ? bq : (mat==1) ? bk : bv;
  float*       out  = (mat==0) ? xq : (mat==1) ? xk : xv;
  const int n0 = nt*16;
  const int lr = lane & 15, kh = lane >> 4;

  // Hoist B fragment + bias (constant for this wave's combo)
  float br0[16], br1[16];
#pragma unroll
  for (int k = 0; k < 16; ++k) {
    const float* bptr = W + (size_t)(k*4 + kh*2)*C + n0 + lr;
    br0[k] = bptr[0];
    br1[k] = bptr[C];
  }
  const float bb = bias[n0+lr];

  for (int mt = wave / 12; mt < NPTS/16; mt += QKV_WAVES/12) {
    const float* arow = x + (size_t)(mt*16+lr)*C + kh*2;
    v8f acc = {};
#pragma unroll
    for (int k = 0; k < 16; ++k) {
      v2f a = *(const v2f*)(arow + k*4);
      v2f b; b[0] = br0[k]; b[1] = br1[k];
      acc = __builtin_amdgcn_wmma_f32_16x16x4_f32(false, a, false, b,
                                                  (short)0, acc, false, false);
    }
#pragma unroll
    for (int r = 0; r < 8; ++r)
      out[(size_t)(mt*16 + r + kh*8)*C + n0 + lr] = acc[r] + bb;
  }
}

// ---------------------------------------------------------------------------
// Kernel 2: sum / sumsq of (rel_p @ Wp1 + bp1) per 3 channels -> pstats[6]
// ---------------------------------------------------------------------------
__global__ void __launch_bounds__(256) k_pstats(
    const float* __restrict__ p, const int* __restrict__ idx,
    const float* __restrict__ Wp1, const float* __restrict__ bp1,
    float* __restrict__ pstats)
{
  __shared__ float sh[6];
  if (threadIdx.x < 6) sh[threadIdx.x] = 0.f;
  __syncthreads();
  float s[3] = {0,0,0}, s2[3] = {0,0,0};
  const int tid = blockIdx.x*blockDim.x + threadIdx.x;
  const int stride = gridDim.x*blockDim.x;
  for (int pr = tid; pr < PAIRS; pr += stride) {
    const int n  = pr >> 4;
    const int id = idx[pr];
    float r0 = p[id*3+0]-p[n*3+0];
    float r1 = p[id*3+1]-p[n*3+1];
    float r2 = p[id*3+2]-p[n*3+2];
#pragma unroll
    for (int k = 0; k < 3; ++k) {
      float t = fmaf(r2, Wp1[6+k], fmaf(r1, Wp1[3+k], fmaf(r0, Wp1[k], bp1[k])));
      s[k] += t; s2[k] += t*t;
    }
  }
#pragma unroll
  for (int k = 0; k < 3; ++k) { atomicAdd(&sh[k], s[k]); atomicAdd(&sh[3+k], s2[k]); }
  __syncthreads();
  if (threadIdx.x < 6) atomicAdd(&pstats[threadIdx.x], sh[threadIdx.x]);
}

// ---------------------------------------------------------------------------
// Kernel 3: sum / sumsq of w = xk_g - xq + pr per 64 channels -> wstats[128]
// wave-per-pair, lane owns channels (lane, lane+32) => coalesced gathers.
// ---------------------------------------------------------------------------
__global__ void __launch_bounds__(256) k_wstats(
    const float* __restrict__ p, const int* __restrict__ idx,
    const float* __restrict__ xq, const float* __restrict__ xk,
    const float* __restrict__ Wp1, const float* __restrict__ bp1,
    const float* __restrict__ gp,  const float* __restrict__ bp_bn,
    const float* __restrict__ Wp2, const float* __restrict__ bp2,
    const float* __restrict__ pstats, float* __restrict__ wstats)
{
  __shared__ float ss[64], ss2[64];
  if (threadIdx.x < 64) { ss[threadIdx.x] = 0.f; ss2[threadIdx.x] = 0.f; }
  __syncthreads();
  const int lane = threadIdx.x & 31;
  const int c = lane, d = lane + 32;
  const BNp B = make_bnp(pstats, gp, bp_bn);
  const float w0c = Wp2[c],   w1c = Wp2[64+c], w2c = Wp2[128+c], bpc = bp2[c];
  const float w0d = Wp2[d],   w1d = Wp2[64+d], w2d = Wp2[128+d], bpd = bp2[d];
  float sc=0.f, s2c=0.f, sd=0.f, s2d=0.f;
  const int wave = (blockIdx.x*blockDim.x + threadIdx.x) >> 5;
  const int nw   = (gridDim.x*blockDim.x) >> 5;
  for (int pr = wave; pr < PAIRS; pr += nw) {
    const int n  = pr >> 4;
    const int id = idx[pr];
    float a0,a1,a2; pfeat(p, n, id, Wp1, bp1, B, a0,a1,a2);
    float prc = fmaf(a2,w2c, fmaf(a1,w1c, fmaf(a0,w0c, bpc)));
    float prd = fmaf(a2,w2d, fmaf(a1,w1d, fmaf(a0,w0d, bpd)));
    float wc = xk[(size_t)id*C+c] - xq[(size_t)n*C+c] + prc;
    float wd = xk[(size_t)id*C+d] - xq[(size_t)n*C+d] + prd;
    sc += wc; s2c += wc*wc; sd += wd; s2d += wd*wd;
  }
  atomicAdd(&ss[c], sc);  atomicAdd(&ss2[c], s2c);
  atomicAdd(&ss[d], sd);  atomicAdd(&ss2[d], s2d);
  __syncthreads();
  if (threadIdx.x < 64) {
    atomicAdd(&wstats[threadIdx.x],    ss[threadIdx.x]);
    atomicAdd(&wstats[64+threadIdx.x], ss2[threadIdx.x]);
  }
}

// ---------------------------------------------------------------------------
// Kernel 4: stats of w2 = relu(BN1(w)) @ Ww1 + bw1 per 8 channels -> mstats[16]
// 64->8 GEMM per pair via wave32 xor-butterfly reduction.
// ---------------------------------------------------------------------------
__global__ void __launch_bounds__(256) k_mstats(
    const float* __restrict__ p, const int* __restrict__ idx,
    const float* __restrict__ xq, const float* __restrict__ xk,
    const float* __restrict__ Wp1, const float* __restrict__ bp1,
    const float* __restrict__ gp,  const float* __restrict__ bp_bn,
    const float* __restrict__ Wp2, const float* __restrict__ bp2,
    const float* __restrict__ gw1, const float* __restrict__ bw1_bn,
    const float* __restrict__ Ww1, const float* __restrict__ bw1,
    const float* __restrict__ pstats, const float* __restrict__ wstats,
    float* __restrict__ mstats)
{
  __shared__ float sm[8], sm2[8];
  if (threadIdx.x < 8) { sm[threadIdx.x] = 0.f; sm2[threadIdx.x] = 0.f; }
  __syncthreads();
  const int lane = threadIdx.x & 31;
  const int c = lane, d = lane + 32;
  const BNp B = make_bnp(pstats, gp, bp_bn);
  const float invM = 1.0f/(float)PAIRS;
  float m1c = wstats[c]*invM;
  float s1c = rsqrtf(wstats[64+c]*invM - m1c*m1c + EPSF)*gw1[c];
  float b1c = bw1_bn[c];
  float m1d = wstats[d]*invM;
  float s1d = rsqrtf(wstats[64+d]*invM - m1d*m1d + EPSF)*gw1[d];
  float b1d = bw1_bn[d];
  const float w0c = Wp2[c], w1c = Wp2[64+c], w2c = Wp2[128+c], bpc = bp2[c];
  const float w0d = Wp2[d], w1d = Wp2[64+d], w2d = Wp2[128+d], bpd = bp2[d];
  float wa[8], wb[8], bwv[8];
#pragma unroll
  for (int o = 0; o < 8; ++o) { wa[o] = Ww1[c*8+o]; wb[o] = Ww1[d*8+o]; bwv[o] = bw1[o]; }
  float ms[8] = {0,0,0,0,0,0,0,0}, ms2[8] = {0,0,0,0,0,0,0,0};
  const int wave = (blockIdx.x*blockDim.x + threadIdx.x) >> 5;
  const int nw   = (gridDim.x*blockDim.x) >> 5;
  for (int pr = wave; pr < PAIRS; pr += nw) {
    const int n  = pr >> 4;
    const int id = idx[pr];
    float a0,a1,a2; pfeat(p, n, id, Wp1, bp1, B, a0,a1,a2);
    float prc = fmaf(a2,w2c, fmaf(a1,w1c, fmaf(a0,w0c, bpc)));
    float prd = fmaf(a2,w2d, fmaf(a1,w1d, fmaf(a0,w0d, bpd)));
    float wc = xk[(size_t)id*C+c] - xq[(size_t)n*C+c] + prc;
    float wd = xk[(size_t)id*C+d] - xq[(size_t)n*C+d] + prd;
    float hc = fmaxf(fmaf(wc-m1c, s1c, b1c), 0.f);
    float hd = fmaxf(fmaf(wd-m1d, s1d, b1d), 0.f);
    float part[8];
#pragma unroll
    for (int o = 0; o < 8; ++o) part[o] = fmaf(hd, wb[o], hc*wa[o]);
#pragma unroll
    for (int o = 0; o < 8; ++o)
#pragma unroll
      for (int off = 16; off >= 1; off >>= 1)
        part[o] += __shfl_xor(part[o], off, 32);
#pragma unroll
    for (int o = 0; o < 8; ++o) {
      float w2 = part[o] + bwv[o];
      ms[o] += w2; ms2[o] += w2*w2;
    }
  }
  if (lane == 0) {
#pragma unroll
    for (int o = 0; o < 8; ++o) { atomicAdd(&sm[o], ms[o]); atomicAdd(&sm2[o], ms2[o]); }
  }
  __syncthreads();
  if (threadIdx.x < 8) {
    atomicAdd(&mstats[threadIdx.x],   sm[threadIdx.x]);
    atomicAdd(&mstats[8+threadIdx.x], sm2[threadIdx.x]);
  }
}

// ---------------------------------------------------------------------------
// Kernel 5: fused attention + softmax + aggregation. One wave per point.
// ---------------------------------------------------------------------------
__global__ void __launch_bounds__(256) k_final(
    const float* __restrict__ p, const int* __restrict__ idx,
    const float* __restrict__ xq, const float* __restrict__ xk,
    const float* __restrict__ xv,
    const float* __restrict__ Wp1, const float* __restrict__ bp1,
    const float* __restrict__ gp,  const float* __restrict__ bp_bn,
    const float* __restrict__ Wp2, const float* __restrict__ bp2,
    const float* __restrict__ gw1, const float* __restrict__ bw1_bn,
    const float* __restrict__ Ww1, const float* __restrict__ bw1,
    const float* __restrict__ gw2, const float* __restrict__ bw2_bn,
    const float* __restrict__ Ww2, const float* __restrict__ bw2,
    const float* __restrict__ pstats, const float* __restrict__ wstats,
    const float* __restrict__ mstats, float* __restrict__ out)
{
  __shared__ float sW2[64];
  __shared__ float sB2[8];
  __shared__ float shw3[8][NS*CH];
  __shared__ float sden[8][CH];
  if (threadIdx.x < 64) sW2[threadIdx.x] = Ww2[threadIdx.x];
  if (threadIdx.x < 8)  sB2[threadIdx.x] = bw2[threadIdx.x];
  __syncthreads();

  const int lane  = threadIdx.x & 31;
  const int wslot = threadIdx.x >> 5;
  const int n     = (blockIdx.x*blockDim.x + threadIdx.x) >> 5;  // point id
  float* lw = shw3[wslot];
  const int c = lane, d = lane + 32;

  const BNp B = make_bnp(pstats, gp, bp_bn);
  const float invM = 1.0f/(float)PAIRS;
  float m1c = wstats[c]*invM;
  float s1c = rsqrtf(wstats[64+c]*invM - m1c*m1c + EPSF)*gw1[c];
  float b1c = bw1_bn[c];
  float m1d = wstats[d]*invM;
  float s1d = rsqrtf(wstats[64+d]*invM - m1d*m1d + EPSF)*gw1[d];
  float b1d = bw1_bn[d];
  const float w0c = Wp2[c], w1c = Wp2[64+c], w2c = Wp2[128+c], bpc = bp2[c];
  const float w0d = Wp2[d], w1d = Wp2[64+d], w2d = Wp2[128+d], bpd = bp2[d];
  float wa[8], wb[8], bwv[8], m2[8], s2v[8], b2v[8];
#pragma unroll
  for (int o = 0; o < 8; ++o) {
    wa[o] = Ww1[c*8+o]; wb[o] = Ww1[d*8+o]; bwv[o] = bw1[o];
    float m = mstats[o]*invM;
    m2[o]  = m;
    s2v[o] = rsqrtf(mstats[8+o]*invM - m*m + EPSF)*gw2[o];
    b2v[o] = bw2_bn[o];
  }

  // Neighbor ids are wave-uniform: load the 16 once, reuse in all phases.
  int nid[NS];
#pragma unroll
  for (int j = 0; j < NS; ++j) nid[j] = idx[n*NS + j];

  // Phase 1: attention logits w3[j][t] -> per-wave LDS
  for (int j = 0; j < NS; ++j) {
    const int id = nid[j];
    // warm WGP$/L2 line for the phase-3 xv gather (global_prefetch_b8)
    __builtin_prefetch(&xv[(size_t)id*C + c], 0, 1);
    float a0,a1,a2; pfeat(p, n, id, Wp1, bp1, B, a0,a1,a2);
    float prc = fmaf(a2,w2c, fmaf(a1,w1c, fmaf(a0,w0c, bpc)));
    float prd = fmaf(a2,w2d, fmaf(a1,w1d, fmaf(a0,w0d, bpd)));
    float wc = xk[(size_t)id*C+c] - xq[(size_t)n*C+c] + prc;
    float wd = xk[(size_t)id*C+d] - xq[(size_t)n*C+d] + prd;
    float hc = fmaxf(fmaf(wc-m1c, s1c, b1c), 0.f);
    float hd = fmaxf(fmaf(wd-m1d, s1d, b1d), 0.f);
    float part[8];
#pragma unroll
    for (int o = 0; o < 8; ++o) part[o] = fmaf(hd, wb[o], hc*wa[o]);
#pragma unroll
    for (int o = 0; o < 8; ++o)
#pragma unroll
      for (int off = 16; off >= 1; off >>= 1)
        part[o] += __shfl_xor(part[o], off, 32);
    float h2[8];
#pragma unroll
    for (int o = 0; o < 8; ++o) {
      float w2 = part[o] + bwv[o];
      h2[o] = fmaxf(fmaf(w2 - m2[o], s2v[o], b2v[o]), 0.f);
    }
    if (lane == 0) {
#pragma unroll
      for (int t = 0; t < 8; ++t) {
        float a = sB2[t];
#pragma unroll
        for (int o = 0; o < 8; ++o) a = fmaf(h2[o], sW2[o*8+t], a);
        lw[j*8 + t] = a;
      }
    }
  }
  __syncthreads();

  // Phase 2: softmax over NS for each of the 8 shared planes
  if (lane < 8) {
    float mx = -3.4e38f;
    for (int j = 0; j < NS; ++j) mx = fmaxf(mx, lw[j*8+lane]);
    float s = 0.f;
    for (int j = 0; j < NS; ++j) {
      float e = __expf(lw[j*8+lane] - mx);
      lw[j*8+lane] = e; s += e;
    }
    sden[wslot][lane] = 1.0f / s;
  }
  __syncthreads();

  // Phase 3: out[n,c] = sum_j (xv_g + pr) * softmax_weight
  const float invd = sden[wslot][lane & 7];
  float ac = 0.f, ad = 0.f;
#pragma unroll 4
  for (int j = 0; j < NS; ++j) {
    const int id = nid[j];
    float a0,a1,a2; pfeat(p, n, id, Wp1, bp1, B, a0,a1,a2);
    float prc = fmaf(a2,w2c, fmaf(a1,w1c, fmaf(a0,w0c, bpc)));
    float prd = fmaf(a2,w2d, fmaf(a1,w1d, fmaf(a0,w0d, bpd)));
    float wgt = lw[j*8 + (lane & 7)] * invd;
    ac = fmaf(xv[(size_t)id*C+c] + prc, wgt, ac);
    ad = fmaf(xv[(size_t)id*C+d] + prd, wgt, ad);
  }
  out[(size_t)n*C + c] = ac;
  out[(size_t)n*C + d] = ad;
}

// ---------------------------------------------------------------------------
// Host launcher
// ---------------------------------------------------------------------------
extern "C" void kernel_launch(void* const* d_in, const int* in_sizes, int n_in,
                              void* d_out, int out_size, void* d_ws, size_t ws_size,
                              hipStream_t stream)
{
  (void)in_sizes; (void)n_in; (void)out_size; (void)ws_size;
  const float* p     = (const float*)d_in[0];
  const float* x     = (const float*)d_in[1];
  const int*   idx   = (const int*)  d_in[2];
  const float* Wq    = (const float*)d_in[3];
  const float* bq    = (const float*)d_in[4];
  const float* Wk    = (const float*)d_in[5];
  const float* bk    = (const float*)d_in[6];
  const float* Wv    = (const float*)d_in[7];
  const float* bv    = (const float*)d_in[8];
  const float* Wp1   = (const float*)d_in[9];
  const float* bp1   = (const float*)d_in[10];
  const float* gp    = (const float*)d_in[11];
  const float* bp_bn = (const float*)d_in[12];
  const float* Wp2   = (const float*)d_in[13];
  const float* bp2   = (const float*)d_in[14];
  const float* gw1   = (const float*)d_in[15];
  const float* bw1bn = (const float*)d_in[16];
  const float* Ww1   = (const float*)d_in[17];
  const float* bw1   = (const float*)d_in[18];
  const float* gw2   = (const float*)d_in[19];
  const float* bw2bn = (const float*)d_in[20];
  const float* Ww2   = (const float*)d_in[21];
  const float* bw2   = (const float*)d_in[22];

  float* ws     = (float*)d_ws;          // 48 MB + 600 B workspace
  float* xq     = ws;
  float* xk     = xq + (size_t)NPTS*C;
  float* xv     = xk + (size_t)NPTS*C;
  float* pstats = xv + (size_t)NPTS*C;   // 6
  float* wstats = pstats + 6;            // 128
  float* mstats = wstats + 128;          // 16

  hipMemsetAsync(pstats, 0, (6+128+16)*sizeof(float), stream);

  // 1) QKV projections: 12288 waves, 8 waves/block, B fragments register-resident
  k_qkv<<<QKV_WAVES/8, 256, 0, stream>>>(x, Wq,bq, Wk,bk, Wv,bv, xq, xk, xv);
  // 2) positional-branch BN stats
  k_pstats<<<1024, 256, 0, stream>>>(p, idx, Wp1, bp1, pstats);
  // 3) attention-logit BN stats (64 ch)
  k_wstats<<<2048, 256, 0, stream>>>(p, idx, xq, xk, Wp1,bp1, gp,bp_bn, Wp2,bp2,
                                     pstats, wstats);
  // 4) mid BN stats (8 ch)
  k_mstats<<<2048, 256, 0, stream>>>(p, idx, xq, xk, Wp1,bp1, gp,bp_bn, Wp2,bp2,
                                     gw1,bw1bn, Ww1,bw1, pstats, wstats, mstats);
  // 5) fused softmax + aggregation, one wave per point
  k_final<<<NPTS/8, 256, 0, stream>>>(p, idx, xq, xk, xv,
                                      Wp1,bp1, gp,bp_bn, Wp2,bp2,
                                      gw1,bw1bn, Ww1,bw1, gw2,bw2bn, Ww2,bw2,
                                      pstats, wstats, mstats, (float*)d_out);
}